// SSIM_5188320494149
// MI455X (gfx1250) — compile-verified
//
#include <hip/hip_runtime.h>

typedef __attribute__((ext_vector_type(2))) float v2f;
typedef __attribute__((ext_vector_type(8))) float v8f;

#define NVOX   192
#define TS     16
#define HALO   5
#define INW    26           // TS + 2*HALO
#define INP    28           // padded K extent (multiple of 4 for 7 K-chunks)
#define KW     11
#define NTHREADS 512
#define NWAVES 16
#define NBLK   (12*12*12)   // (192/16)^3 = 1728

// ---------------------------------------------------------------------------
// Band-matrix entry: A[row m][col kg] = g1[kg - m] if 0 <= kg-m < 11 else 0.
// out[m] = sum_t g1[t - m] * in[t]  (tile input origin = output origin - HALO)
__device__ __forceinline__ float bandw(const float* g1, int m, int kg) {
  const int d = kg - m;
  return (d >= 0 && d < KW) ? g1[d] : 0.f;
}

// ---------------------------------------------------------------------------
// One 1D conv pass over P pencils: Out(16 outputs x 16 pencils) =
// Band(16x28) * X(28x16) as 7 chained V_WMMA_F32_16X16X4_F32.
// VGPR layouts per CDNA5 ISA 7.12.2 (fp32):
//   A 16x4 : lane L holds A[M=L&15][K = 2*(L>>4) + v]
//   B 4x16 : lane L holds B[K = 2*(L>>4) + v][N = L&15]
//   D 16x16: elem r: lanes 0-15 -> (M=r, N=lane), lanes 16-31 -> (M=r+8, N=lane&15)
// src2(kg, p) returns {X[kg][p], X[kg+1][p]}; all kg in [0,28) must be
// readable (padded, finite).  Out-of-range pencils are handled by clamping the
// load address (their D columns are never stored).
template <class FSrc2, class FDst>
__device__ __forceinline__ void conv_pass(FSrc2 src2, FDst dstW, int P,
                                          const v2f* aband) {
  const int lane = threadIdx.x & 31;
  const int wave = threadIdx.x >> 5;
  const int hi   = lane >> 4;
  const int m    = lane & 15;
  const int numTiles = (P + 15) >> 4;
  for (int tile = wave; tile < numTiles; tile += NWAVES) {  // uniform per wave
    const int p  = (tile << 4) + m;
    const int pl = (p < P) ? p : (P - 1);    // clamped load pencil (finite data)
    v2f b[7];
#pragma unroll
    for (int chunk = 0; chunk < 7; ++chunk)
      b[chunk] = src2(chunk * 4 + hi * 2, pl);
    v8f c = {};
#pragma unroll
    for (int chunk = 0; chunk < 7; ++chunk)
      c = __builtin_amdgcn_wmma_f32_16x16x4_f32(false, aband[chunk],
                                                false, b[chunk],
                                                (short)0, c, false, false);
    if (p < P) {
#pragma unroll
      for (int r = 0; r < 8; ++r) dstW(r + hi * 8, p, c[r]);
    }
  }
}

// ---------------------------------------------------------------------------
// One field F through all three separable passes.
// F: 0=gr, 1=gt, 2=gr*gr, 3=gt*gt (accumulated into dstR), 4=gr*gt
// Layouts (all K-dims padded to 28, pads zero-filled):
//   sA/sB : [z26][y26][x28]            off = (z*26+y)*28 + x
//   t1    : [z26][y28][xo16]           off = z*448 + y*16 + xo
//   t2    : [z28][yo16][xo16]          off = z*256 + yo*16 + xo
template <int F>
__device__ __forceinline__ void do_field(const float* sA, const float* sB,
                                         float* t1, float* t2, float* dstR,
                                         const v2f* aband) {
  // Pass X: pencil p = z*26+y (P=676), conv along x -> t1
  auto srcX = [&](int kg, int p) -> v2f {
    const int off = p * INP + kg;              // 8B aligned (kg even, INP even)
    if constexpr (F == 0) return *(const v2f*)(sA + off);
    else if constexpr (F == 1) return *(const v2f*)(sB + off);
    else if constexpr (F == 2) { const v2f v = *(const v2f*)(sA + off); return v * v; }
    else if constexpr (F == 3) { const v2f v = *(const v2f*)(sB + off); return v * v; }
    else return *(const v2f*)(sA + off) * *(const v2f*)(sB + off);
  };
  auto dstX = [&](int to, int p, float v) {
    const int z = p / INW, y = p - z * INW;
    t1[z * (INP * TS) + y * TS + to] = v;
  };
  conv_pass(srcX, dstX, INW * INW, aband);
  __syncthreads();

  // Pass Y: pencil q = z*16+x (P=416, multiple of 16), conv along y -> t2
  auto srcY = [&](int kg, int q) -> v2f {
    const int base = (q >> 4) * (INP * TS) + kg * TS + (q & 15);
    v2f r; r.x = t1[base]; r.y = t1[base + TS]; return r;
  };
  auto dstY = [&](int to, int q, float v) {
    t2[(q >> 4) * (TS * TS) + to * TS + (q & 15)] = v;
  };
  conv_pass(srcY, dstY, INW * TS, aband);
  __syncthreads();

  // Pass Z: pencil p = y*16+x (P=256, multiple of 16), conv along z -> result
  auto srcZ = [&](int kg, int p) -> v2f {
    const int base = kg * (TS * TS) + p;
    v2f r; r.x = t2[base]; r.y = t2[base + TS * TS]; return r;
  };
  auto dstZ = [&](int to, int p, float v) {
    const int off = to * (TS * TS) + p;
    if constexpr (F == 3) dstR[off] += v; else dstR[off] = v;
  };
  conv_pass(srcZ, dstZ, TS * TS, aband);
  __syncthreads();
}

// ---------------------------------------------------------------------------
__global__ __launch_bounds__(NTHREADS, 1)
void ssim_main(const float* __restrict__ gr, const float* __restrict__ gt,
               const float* __restrict__ window, float* __restrict__ partials) {
  __shared__ float sA[INW * INW * INP];     // 75.7 KB  gr tile (rescaled, x-padded)
  __shared__ float sB[INW * INW * INP];     // 75.7 KB  gt tile
  __shared__ float t1[INW * INP * TS];      // 46.6 KB  (y padded to 28)
  __shared__ float t2[INP * TS * TS];       // 28.7 KB  (z padded to 28)
  __shared__ float rMu1[TS * TS * TS];      // 16 KB
  __shared__ float rMu2[TS * TS * TS];      // 16 KB
  __shared__ float rAcc[TS * TS * TS];      // 16 KB  conv(x^2)+conv(y^2)
  __shared__ float rC12[TS * TS * TS];      // 16 KB
  __shared__ float g1[16];
  __shared__ float redf[NTHREADS];          // ~295 KB total < 320 KB/WGP

  // 1D weights = marginals of the (sum==1) separable 3D window
  if (threadIdx.x < 16) {
    float s = 0.f;
    if (threadIdx.x < KW) {
      const float* w = window + threadIdx.x * (KW * KW);
      for (int i = 0; i < KW * KW; ++i) s += w[i];
    }
    g1[threadIdx.x] = s;
  }

  // Zero the K-padding rows of t1 (y=26,27) and t2 (z=26,27)
  for (int i = threadIdx.x; i < INW * 2 * TS; i += NTHREADS) {
    const int z = i / (2 * TS), r = i % (2 * TS);
    t1[z * (INP * TS) + (INW + r / TS) * TS + (r % TS)] = 0.f;
  }
  for (int i = threadIdx.x; i < 2 * TS * TS; i += NTHREADS)
    t2[INW * (TS * TS) + i] = 0.f;

  // Load halo'd input tiles with (x+1)/2 rescale; zero-pad outside + x-pad
  const int bx = (int)blockIdx.x * TS - HALO;
  const int by = (int)blockIdx.y * TS - HALO;
  const int bz = (int)blockIdx.z * TS - HALO;
  for (int idx = threadIdx.x; idx < INW * INW * INP; idx += NTHREADS) {
    const int x = idx % INP, rest = idx / INP;
    const int y = rest % INW, z = rest / INW;
    const int gx = bx + x, gy = by + y, gz = bz + z;
    float a = 0.f, b = 0.f;
    if (x < INW && (unsigned)gx < NVOX && (unsigned)gy < NVOX && (unsigned)gz < NVOX) {
      const size_t g = ((size_t)gz * NVOX + gy) * NVOX + gx;
      a = (gr[g] + 1.f) * 0.5f;
      b = (gt[g] + 1.f) * 0.5f;
    }
    sA[idx] = a;
    sB[idx] = b;
  }
  __syncthreads();

  // Hoist the band matrix A into registers: invariant across tiles & passes
  const int lane = threadIdx.x & 31;
  const int hi   = lane >> 4;
  const int m    = lane & 15;
  v2f aband[7];
#pragma unroll
  for (int chunk = 0; chunk < 7; ++chunk) {
    const int kg = chunk * 4 + hi * 2;
    aband[chunk].x = bandw(g1, m, kg);
    aband[chunk].y = bandw(g1, m, kg + 1);
  }

  do_field<0>(sA, sB, t1, t2, rMu1, aband);   // mu1
  do_field<1>(sA, sB, t1, t2, rMu2, aband);   // mu2
  do_field<2>(sA, sB, t1, t2, rAcc, aband);   // conv(gr^2)
  do_field<3>(sA, sB, t1, t2, rAcc, aband);   // += conv(gt^2)
  do_field<4>(sA, sB, t1, t2, rC12, aband);   // conv(gr*gt)

  // SSIM map + block partial sum
  float sum = 0.f;
  for (int i = threadIdx.x; i < TS * TS * TS; i += NTHREADS) {
    const float mu1 = rMu1[i], mu2 = rMu2[i];
    const float m11 = mu1 * mu1, m22 = mu2 * mu2, m12 = mu1 * mu2;
    const float sAB = rAcc[i] - m11 - m22;   // sigma1_sq + sigma2_sq
    const float s12 = rC12[i] - m12;
    const float C1 = 1e-4f, C2 = 9e-4f;
    const float num = (2.f * m12 + C1) * (2.f * s12 + C2);
    const float den = (m11 + m22 + C1) * (sAB + C2);
    sum += num / den;
  }
  redf[threadIdx.x] = sum;
  __syncthreads();
  for (int off = NTHREADS / 2; off > 0; off >>= 1) {
    if (threadIdx.x < off) redf[threadIdx.x] += redf[threadIdx.x + off];
    __syncthreads();
  }
  if (threadIdx.x == 0) {
    partials[((int)blockIdx.z * gridDim.y + blockIdx.y) * gridDim.x + blockIdx.x] = redf[0];
  }
}

__global__ __launch_bounds__(256, 1)
void ssim_finalize(const float* __restrict__ partials, float* __restrict__ out) {
  __shared__ double buf[256];
  double s = 0.0;
  for (int i = threadIdx.x; i < NBLK; i += 256) s += (double)partials[i];
  buf[threadIdx.x] = s;
  __syncthreads();
  for (int off = 128; off > 0; off >>= 1) {
    if (threadIdx.x < off) buf[threadIdx.x] += buf[threadIdx.x + off];
    __syncthreads();
  }
  if (threadIdx.x == 0) {
    out[0] = (float)(1.0 - buf[0] / (double)((long long)NVOX * NVOX * NVOX));
  }
}

extern "C" void kernel_launch(void* const* d_in, const int* in_sizes, int n_in,
                              void* d_out, int out_size, void* d_ws, size_t ws_size,
                              hipStream_t stream) {
  (void)in_sizes; (void)n_in; (void)out_size; (void)ws_size;
  const float* gr     = (const float*)d_in[0];
  const float* gt     = (const float*)d_in[1];
  const float* window = (const float*)d_in[2];
  float* partials     = (float*)d_ws;   // 1728 floats of scratch

  dim3 grid(NVOX / TS, NVOX / TS, NVOX / TS);
  ssim_main<<<grid, NTHREADS, 0, stream>>>(gr, gt, window, partials);
  ssim_finalize<<<1, 256, 0, stream>>>(partials, (float*)d_out);
}